// QuanvolutionKernelClassifier_65481071400357
// MI455X (gfx1250) — compile-verified
//
#include <hip/hip_runtime.h>
#include <math.h>

// Quanvolution classifier for MI455X (gfx1250, wave32).
// feats(b, p*196+q) = |prod_i cos(0.5*(x[b,patch q,i] - proto[p,i]))|
// logits = feats @ W^T + b ; out = log_softmax(logits)
//
// Strategy:
//  * cos(0.5(x-y)) = cos(.5x)cos(.5y) + sin(.5x)sin(.5y): per-pixel cos/sin
//    computed once per (row,patch), prototype cos/sin cached in registers.
//  * GEMM over K=1960 reordered as (patch, proto) with protos padded to 12,
//    done as 588 chunks of K=4 via V_WMMA_F32_16X16X4_F32 per 16-row tile.
//  * W staged in LDS (78.4 KB); B fragments read per chunk.
//  * log-softmax via width-16 shuffle butterflies (rows live across 16 lanes).

typedef float v2f __attribute__((ext_vector_type(2)));
typedef float v8f __attribute__((ext_vector_type(8)));

#define BATCH   8192
#define NPROTO  10
#define NCLASS  10
#define NPATCH  196
#define KFEAT   (NPROTO * NPATCH)   // 1960
#define ROWS_PER_WAVE 16
#define WAVES_PER_BLOCK 4
#define ROWS_PER_BLOCK (ROWS_PER_WAVE * WAVES_PER_BLOCK)  // 64

__launch_bounds__(32 * WAVES_PER_BLOCK)
__global__ void quanv_wmma_kernel(const float* __restrict__ x,
                                  const float* __restrict__ protos,
                                  const float* __restrict__ W,
                                  const float* __restrict__ bias,
                                  float* __restrict__ out)
{
    __shared__ float ldsW[NCLASS * KFEAT];   // 19600 floats = 78.4 KB

    const int tid = threadIdx.x;

    // Cooperative stage of W into LDS (fully cached by L2 anyway; LDS gives
    // deterministic low-latency B-fragment gathers).
    for (int i = tid; i < NCLASS * KFEAT; i += blockDim.x) ldsW[i] = W[i];
    __syncthreads();

    const int lane = tid & 31;
    const int wave = tid >> 5;
    const int hi   = (lane >> 4) & 1;   // 0: lanes 0-15 (K 0,1), 1: lanes 16-31 (K 2,3)
    const int n    = lane & 15;         // A row index / B column index for this lane
    const int row0 = blockIdx.x * ROWS_PER_BLOCK + wave * ROWS_PER_WAVE;
    const int row  = row0 + n;

    // Prototype constants this lane will ever need:
    //   chunk j (j=0..2) uses protos p0 = 4j + 2*hi, p1 = p0+1  (p in [0,12))
    // cos/sin of 0.5*proto, 6 protos x 4 components, register resident.
    float cy[6][4], sy[6][4];
#pragma unroll
    for (int j = 0; j < 3; ++j) {
#pragma unroll
        for (int t = 0; t < 2; ++t) {
            int p   = 4 * j + 2 * hi + t;
            int pcl = (p < NPROTO) ? p : 0;   // clamp pad protos (W side is zeroed)
#pragma unroll
            for (int i = 0; i < 4; ++i) {
                float y = 0.5f * protos[pcl * 4 + i];
                cy[j * 2 + t][i] = __cosf(y);
                sy[j * 2 + t][i] = __sinf(y);
            }
        }
    }

    v8f acc = {0.f, 0.f, 0.f, 0.f, 0.f, 0.f, 0.f, 0.f};

    const float* xr = x + (size_t)row * 784;
    const bool okn = (n < NCLASS);

    for (int q = 0; q < NPATCH; ++q) {
        // patch q = pr*14 + pc ; pixels at offsets {0,1,28,29} from 56*pr+2*pc
        int pr  = q / 14;
        int pc  = q - pr * 14;
        int off = 56 * pr + 2 * pc;
        float2 t0 = *(const float2*)(xr + off);
        float2 t1 = *(const float2*)(xr + off + 28);

        float c0 = __cosf(0.5f * t0.x), s0 = __sinf(0.5f * t0.x);
        float c1 = __cosf(0.5f * t0.y), s1 = __sinf(0.5f * t0.y);
        float c2 = __cosf(0.5f * t1.x), s2 = __sinf(0.5f * t1.x);
        float c3 = __cosf(0.5f * t1.y), s3 = __sinf(0.5f * t1.y);

#pragma unroll
        for (int j = 0; j < 3; ++j) {
            const int p0 = 4 * j + 2 * hi;   // this lane's K slice -> protos p0, p0+1

            // A fragment: two feature values (row = n, K = p0 / p0+1 of this patch)
            float f0 = fmaf(s0, sy[2 * j][0], c0 * cy[2 * j][0]);
            f0      *= fmaf(s1, sy[2 * j][1], c1 * cy[2 * j][1]);
            f0      *= fmaf(s2, sy[2 * j][2], c2 * cy[2 * j][2]);
            f0      *= fmaf(s3, sy[2 * j][3], c3 * cy[2 * j][3]);
            float f1 = fmaf(s0, sy[2 * j + 1][0], c0 * cy[2 * j + 1][0]);
            f1      *= fmaf(s1, sy[2 * j + 1][1], c1 * cy[2 * j + 1][1]);
            f1      *= fmaf(s2, sy[2 * j + 1][2], c2 * cy[2 * j + 1][2]);
            f1      *= fmaf(s3, sy[2 * j + 1][3], c3 * cy[2 * j + 1][3]);

            v2f a;
            a.x = fabsf(f0);
            a.y = fabsf(f1);

            // B fragment: W[n][p*196 + q] for p = p0, p0+1 (zero-padded cols/protos)
            const int  base = n * KFEAT + q;
            const bool ok0  = okn && (p0 < NPROTO);
            const bool ok1  = okn && (p0 + 1 < NPROTO);
            float w0 = ldsW[base + (ok0 ? p0 : 0) * NPATCH];
            float w1 = ldsW[base + (ok1 ? (p0 + 1) : 0) * NPATCH];
            v2f bfr;
            bfr.x = ok0 ? w0 : 0.f;
            bfr.y = ok1 ? w1 : 0.f;

            // D = A(16x4) * B(4x16) + C, f32 accumulate
            acc = __builtin_amdgcn_wmma_f32_16x16x4_f32(
                false, a, false, bfr, (short)0, acc, false, false);
        }
    }

    // Bias + log-softmax. Accumulator VGPR r holds row M = r + 8*hi, col n.
    float bl = okn ? bias[n] : 0.f;
#pragma unroll
    for (int r = 0; r < 8; ++r) {
        float logit = acc[r] + bl;
        float v = okn ? logit : -1e30f;
        v = fmaxf(v, __shfl_xor(v, 1, 16));
        v = fmaxf(v, __shfl_xor(v, 2, 16));
        v = fmaxf(v, __shfl_xor(v, 4, 16));
        v = fmaxf(v, __shfl_xor(v, 8, 16));
        float e = okn ? __expf(logit - v) : 0.f;
        float ss = e;
        ss += __shfl_xor(ss, 1, 16);
        ss += __shfl_xor(ss, 2, 16);
        ss += __shfl_xor(ss, 4, 16);
        ss += __shfl_xor(ss, 8, 16);
        float lsm = logit - v - __logf(ss);
        int M = r + hi * 8;
        if (okn) out[(size_t)(row0 + M) * NCLASS + n] = lsm;
    }
}

extern "C" void kernel_launch(void* const* d_in, const int* in_sizes, int n_in,
                              void* d_out, int out_size, void* d_ws, size_t ws_size,
                              hipStream_t stream) {
    const float* x      = (const float*)d_in[0];   // (8192, 784)
    const float* protos = (const float*)d_in[1];   // (10, 4)
    const float* W      = (const float*)d_in[2];   // (10, 1960)
    const float* b      = (const float*)d_in[3];   // (10,)
    float* out          = (float*)d_out;           // (8192, 10)

    dim3 grid(BATCH / ROWS_PER_BLOCK);             // 128 blocks
    dim3 block(32 * WAVES_PER_BLOCK);              // 128 threads = 4 waves
    quanv_wmma_kernel<<<grid, block, 0, stream>>>(x, protos, W, b, out);
    (void)in_sizes; (void)n_in; (void)out_size; (void)d_ws; (void)ws_size;
}